// Dct2d_12438225289794
// MI455X (gfx1250) — compile-verified
//
#include <hip/hip_runtime.h>

// Blockwise 8x8 2D DCT as one GEMM:  Out[64 x B] = (A (x) A)[64x64] @ X[64 x B]
// Memory-bound (512MB traffic, ~8.6 GFLOP) -> floor ~22us @ 23.3 TB/s.
// v3: async global->LDS staging (ASYNCcnt) + double buffering, WMMA f32 16x16x4.

typedef __attribute__((ext_vector_type(2))) float v2f;
typedef __attribute__((ext_vector_type(8))) float v8f;
typedef __attribute__((ext_vector_type(4))) int   v4i;

#define LDS_PAD 68          // 64 + 4 pad floats: A/B ds_load_b64 gathers hit all 64 banks
#define WAVES_PER_WG 4
#define THREADS (WAVES_PER_WG * 32)
#define GRID 4096
#define JOBS_PER_WAVE 4     // 4096 WGs * 4 waves * 4 jobs = 65536 = 64*128*8 jobs

typedef __attribute__((address_space(1))) v4i gv4i;   // global int4
typedef __attribute__((address_space(3))) v4i lv4i;   // LDS int4

// One 16-byte async copy: global (per-lane contiguous) -> LDS (per-lane address).
static __device__ __forceinline__ void async_copy16(const float* g, float* l) {
#if __has_builtin(__builtin_amdgcn_global_load_async_to_lds_b128)
    __builtin_amdgcn_global_load_async_to_lds_b128((gv4i*)g, (lv4i*)l, 0, 0);
#else
    unsigned lds = (unsigned)(unsigned long long)l;   // LDS byte offset = addr[31:0]
    asm volatile("global_load_async_to_lds_b128 %0, %1, off"
                 :: "v"(lds), "v"(g) : "memory");
#endif
}

template <int N>
static __device__ __forceinline__ void wait_async() {
#if __has_builtin(__builtin_amdgcn_s_wait_asynccnt)
    __builtin_amdgcn_s_wait_asynccnt(N);
#else
    if constexpr (N == 8) asm volatile("s_wait_asynccnt 0x8" ::: "memory");
    else                  asm volatile("s_wait_asynccnt 0x0" ::: "memory");
#endif
    asm volatile("" ::: "memory");   // keep the LDS reads below the wait
}

__global__ __launch_bounds__(THREADS)
void dct8x8_wmma_kernel(const float* __restrict__ x,
                        const float* __restrict__ A,
                        float* __restrict__ out)
{
    __shared__ float Ms[64 * LDS_PAD];                       // M64 = A (x) A
    __shared__ float Xs[WAVES_PER_WG][2][16 * LDS_PAD];      // per-wave double buffer

    const int tid  = threadIdx.x;
    const int lane = tid & 31;
    const int wv   = tid >> 5;
    const int lh   = lane >> 4;   // lane half: selects K pair within a 4-chunk
    const int ln   = lane & 15;   // operand row/col id

    // Build M64[ch][k] = A[ch/8][k/8] * A[ch%8][k%8] once per workgroup.
    for (int idx = tid; idx < 4096; idx += THREADS) {
        int row = idx >> 6, col = idx & 63;
        Ms[row * LDS_PAD + col] =
            A[(row >> 3) * 8 + (col >> 3)] * A[(row & 7) * 8 + (col & 7)];
    }
    __syncthreads();

    const long waveId = (long)blockIdx.x * WAVES_PER_WG + wv;
    const int  b  = lane >> 1;            // block index of this lane's 16B
    const int  c0 = (lane & 1) * 4;       // column-within-block (0 or 4)
    float* xsbuf[2] = { &Xs[wv][0][0], &Xs[wv][1][0] };

    // Issue async staging for one job: 8 rows x (32 lanes x 16B) = 4KB strip,
    // transposed on the fly into X[b][k] (k = r*8 + c) layout.
    auto stage = [&](int t, float* xs) {
        long job = waveId * JOBS_PER_WAVE + t;
        int gq = (int)(job & 7);
        int gh = (int)((job >> 3) & 127);
        int nb = (int)(job >> 10);
        const float* src = x + ((long)nb << 20) + (long)(gh * 8) * 1024
                             + gq * 128 + lane * 4;
        float* dst = xs + b * LDS_PAD + c0;
        #pragma unroll
        for (int r = 0; r < 8; ++r)
            async_copy16(src + (long)r * 1024, dst + r * 8);
    };

    stage(0, xsbuf[0]);   // prologue

    #pragma unroll
    for (int t = 0; t < JOBS_PER_WAVE; ++t) {
        float* xs = xsbuf[t & 1];
        if (t + 1 < JOBS_PER_WAVE) {
            stage(t + 1, xsbuf[(t + 1) & 1]);   // prefetch next job
            wait_async<8>();                    // in-order: current job's 8 are done
        } else {
            wait_async<0>();
        }

        // ---- Out[64x16] = M64 @ X : 16 K-chunks x 4 channel groups, WMMA f32 16x16x4.
        // Operand layout (ISA 7.12.2): element(id = lane%16, K = 2*(lane/16) + vgpr).
        v8f acc0 = {}, acc1 = {}, acc2 = {}, acc3 = {};
        #pragma unroll
        for (int kc = 0; kc < 16; ++kc) {
            const int kb = kc * 4 + lh * 2;
            v2f bop = *(const v2f*)(xs + ln * LDS_PAD + kb);             // X slice
            v2f a0  = *(const v2f*)(Ms + (ln     ) * LDS_PAD + kb);      // ch 0..15
            v2f a1  = *(const v2f*)(Ms + (16 + ln) * LDS_PAD + kb);      // ch 16..31
            v2f a2  = *(const v2f*)(Ms + (32 + ln) * LDS_PAD + kb);      // ch 32..47
            v2f a3  = *(const v2f*)(Ms + (48 + ln) * LDS_PAD + kb);      // ch 48..63
            acc0 = __builtin_amdgcn_wmma_f32_16x16x4_f32(false, a0, false, bop,
                                                         (short)0, acc0, false, false);
            acc1 = __builtin_amdgcn_wmma_f32_16x16x4_f32(false, a1, false, bop,
                                                         (short)0, acc1, false, false);
            acc2 = __builtin_amdgcn_wmma_f32_16x16x4_f32(false, a2, false, bop,
                                                         (short)0, acc2, false, false);
            acc3 = __builtin_amdgcn_wmma_f32_16x16x4_f32(false, a3, false, bop,
                                                         (short)0, acc3, false, false);
        }

        // ---- Store. C/D layout: vgpr rr, lane -> ch = g*16 + 8*lh + rr, gw-block = ln.
        long job = waveId * JOBS_PER_WAVE + t;
        int gq = (int)(job & 7);
        int gh = (int)((job >> 3) & 127);
        int nb = (int)(job >> 10);
        float* dst = out + ((long)nb << 20) + (long)(8 * lh) * 16384
                         + gh * 128 + gq * 16 + ln;
        #pragma unroll
        for (int rr = 0; rr < 8; ++rr) {
            dst[(long)(rr     ) * 16384] = acc0[rr];
            dst[(long)(16 + rr) * 16384] = acc1[rr];
            dst[(long)(32 + rr) * 16384] = acc2[rr];
            dst[(long)(48 + rr) * 16384] = acc3[rr];
        }
    }
}

extern "C" void kernel_launch(void* const* d_in, const int* in_sizes, int n_in,
                              void* d_out, int out_size, void* d_ws, size_t ws_size,
                              hipStream_t stream) {
    (void)in_sizes; (void)n_in; (void)d_ws; (void)ws_size; (void)out_size;
    const float* x = (const float*)d_in[0];   // (64,1,1024,1024) f32
    const float* A = (const float*)d_in[1];   // (8,8) f32 DCT matrix
    float* out = (float*)d_out;               // (64,64,128,128) f32
    dct8x8_wmma_kernel<<<GRID, THREADS, 0, stream>>>(x, A, out);
}